// HebbianConv2d_74131135529100
// MI455X (gfx1250) — compile-verified
//
#include <hip/hip_runtime.h>
#include <hip/hip_bf16.h>

// ---------------------------------------------------------------------------
// HebbianConv2d (SOM w/ lateral feedback) for MI455X (gfx1250, wave32, WMMA)
//
// All GEMMs run on the f32 matrix path: V_WMMA_F32_16X16X4_F32.
//   A-frag (16x4 f32, 2 VGPRs): lane L(0-15): K={k0,k0+1}; lane L(16-31): K={k0+2,k0+3}
//   B-frag (4x16 f32, 2 VGPRs): assumed mirror layout: b.x=B[k0+2*half][col], b.y=B[k0+2*half+1][col]
//   C/D (16x16 f32, 8 VGPRs):   c[v] = D[row = v + 8*half][col = lane&15]
// ---------------------------------------------------------------------------

typedef __attribute__((ext_vector_type(2))) float v2f;
typedef __attribute__((ext_vector_type(8))) float v8f;

#define NB   32          // batch
#define NC   64          // in channels
#define HH   32          // input H=W
#define OO   256         // out channels (16x16 SOM)
#define HO   28          // output H=W (valid conv, K=5)
#define NN   25088       // NB*HO*HO
#define DD   1600        // NC*5*5
#define Y4SZ 6422528     // NB*OO*HO*HO

__device__ __forceinline__ v8f wmma_f32(v2f a, v2f b, v8f c) {
  return __builtin_amdgcn_wmma_f32_16x16x4_f32(
      /*neg_a=*/false, a, /*neg_b=*/false, b,
      /*c_mod=*/(short)0, c, /*reuse_a=*/false, /*reuse_b=*/false);
}

// ---------------------------------------------------------------------------
// 1) im2col:  x[B,C,32,32] -> x_unf[N, D], d = c*25 + kh*5 + kw
// ---------------------------------------------------------------------------
__global__ void im2col_k(const float* __restrict__ x, float* __restrict__ xu) {
  int n   = blockIdx.x;                 // 0..NN-1
  int b   = n / (HO * HO);
  int rem = n - b * (HO * HO);
  int oh  = rem / HO;
  int ow  = rem - oh * HO;
  for (int d = threadIdx.x; d < DD; d += 256) {
    int c  = d / 25;
    int kk = d - c * 25;
    int kh = kk / 5;
    int kw = kk - kh * 5;
    xu[(long)n * DD + d] = x[((b * NC + c) * HH + oh + kh) * HH + (ow + kw)];
  }
}

// ---------------------------------------------------------------------------
// 2) forward GEMM: y[N,O] = x_unf @ Wf^T  (K = D = 1600)
//    also scatter into y4[B,O,Ho,Wo]
// ---------------------------------------------------------------------------
__global__ void fwd_k(const float* __restrict__ xu, const float* __restrict__ W,
                      float* __restrict__ y, float* __restrict__ y4) {
  int lane = threadIdx.x & 31;
  int wid  = threadIdx.x >> 5;
  int tile = blockIdx.x * 8 + wid;      // 25088 tiles = 1568 (N) x 16 (O)
  int tO   = tile & 15;
  int tN   = tile >> 4;
  int half = lane >> 4;
  int l16  = lane & 15;

  const float* ap = xu + (long)(tN * 16 + l16) * DD + 2 * half;
  const float* bp = W  + (long)(tO * 16 + l16) * DD + 2 * half;

  v8f c = {};
#pragma unroll 4
  for (int k = 0; k < DD; k += 4) {
    v2f a = *(const v2f*)ap;            // x_unf[row][k+2h : k+2h+1]
    v2f b = *(const v2f*)bp;            // Wf[o][k+2h : k+2h+1]  (B = Wf^T)
    c = wmma_f32(a, b, c);
    ap += 4; bp += 4;
  }

#pragma unroll
  for (int v = 0; v < 8; ++v) {
    int n   = tN * 16 + v + 8 * half;
    int o   = tO * 16 + l16;
    float val = c[v];
    y[(long)n * OO + o] = val;
    int b   = n / (HO * HO);
    int rem = n - b * (HO * HO);
    int oh  = rem / HO;
    int ow  = rem - oh * HO;
    y4[((b * OO + o) * HO + oh) * HO + ow] = val;
  }
}

// ---------------------------------------------------------------------------
// 3) winner-take-all + lateral feedback (Gaussian of chessboard distance),
//    r = clip(lfb,-1,1) * y, written IN PLACE over y. One block per site n.
// ---------------------------------------------------------------------------
__global__ void lfb_k(float* __restrict__ yr) {
  __shared__ float red[256];
  __shared__ float sw[256];
  __shared__ float gt[16];
  int n = blockIdx.x;
  int o = threadIdx.x;

  float v = yr[(long)n * OO + o];
  red[o] = v;
  if (o < 16) gt[o] = expf(-(float)(o * o) * (1.0f / 98.0f)); // 2*sigma^2 = 98
  __syncthreads();

  for (int s = 128; s > 0; s >>= 1) {
    if (o < s) red[o] = fmaxf(red[o], red[o + s]);
    __syncthreads();
  }
  float m = red[0];
  sw[o] = (v == m) ? 1.0f : 0.0f;       // exact-equality winner mask (ties kept)
  __syncthreads();

  int oi = o >> 4, oj = o & 15;
  float lfb = 0.0f;
  for (int w = 0; w < 256; ++w) {       // fixed order -> deterministic
    if (sw[w] != 0.0f) {
      int wi = w >> 4, wj = w & 15;
      int di = oi - wi; if (di < 0) di = -di;
      int dj = oj - wj; if (dj < 0) dj = -dj;
      lfb += gt[di > dj ? di : dj];
    }
  }
  lfb = fminf(lfb, 1.0f);               // lfb >= 0, so clip(-1,1) == min(.,1)
  yr[(long)n * OO + o] = lfb * v;       // r
}

// ---------------------------------------------------------------------------
// 4) A = (r^T @ x_unf) / N      [O, D], K = N = 25088
// ---------------------------------------------------------------------------
__global__ void agemm_k(const float* __restrict__ r, const float* __restrict__ xu,
                        float* __restrict__ Aw) {
  int lane = threadIdx.x & 31;
  int wid  = threadIdx.x >> 5;
  int tile = blockIdx.x * 8 + wid;      // 1600 tiles = 16 (O) x 100 (D)
  int tO   = tile / 100;
  int tD   = tile - tO * 100;
  int half = lane >> 4;
  int l16  = lane & 15;

  const float* ap = r  + (long)(2 * half) * OO + tO * 16 + l16;  // A[m][k]=r[k][o]
  const float* bp = xu + (long)(2 * half) * DD + tD * 16 + l16;  // B[k][n]=xu[k][d]

  v8f c = {};
  for (int k = 0; k < NN; k += 4) {
    v2f a; a.x = ap[0]; a.y = ap[OO];
    v2f b; b.x = bp[0]; b.y = bp[DD];
    __builtin_prefetch(bp + 16 * DD, 0, 1);   // global_prefetch_b8 down-stream
    c = wmma_f32(a, b, c);
    ap += 4 * OO; bp += 4 * DD;
  }

  const float invN = 1.0f / (float)NN;
#pragma unroll
  for (int v = 0; v < 8; ++v) {
    int o = tO * 16 + v + 8 * half;
    int d = tD * 16 + l16;
    Aw[(long)o * DD + d] = c[v] * invN;
  }
}

// ---------------------------------------------------------------------------
// 5) M = tril(r^T @ r) / N      [O, O], K = N
// ---------------------------------------------------------------------------
__global__ void mgemm_k(const float* __restrict__ r, float* __restrict__ Mm) {
  int lane = threadIdx.x & 31;
  int wid  = threadIdx.x >> 5;
  int tile = blockIdx.x * 8 + wid;      // 256 tiles = 16 x 16
  int tR   = tile >> 4;
  int tC   = tile & 15;
  int half = lane >> 4;
  int l16  = lane & 15;

  if (tR < tC) {                        // strictly above diagonal: all zeros
#pragma unroll
    for (int v = 0; v < 8; ++v) {
      int i = tR * 16 + v + 8 * half;
      int j = tC * 16 + l16;
      Mm[i * OO + j] = 0.0f;
    }
    return;                             // wave-uniform exit, no barriers used
  }

  const float* ap = r + (long)(2 * half) * OO + tR * 16 + l16;
  const float* bp = r + (long)(2 * half) * OO + tC * 16 + l16;

  v8f c = {};
  for (int k = 0; k < NN; k += 4) {
    v2f a; a.x = ap[0]; a.y = ap[OO];
    v2f b; b.x = bp[0]; b.y = bp[OO];
    c = wmma_f32(a, b, c);
    ap += 4 * OO; bp += 4 * OO;
  }

  const float invN = 1.0f / (float)NN;
#pragma unroll
  for (int v = 0; v < 8; ++v) {
    int i = tR * 16 + v + 8 * half;
    int j = tC * 16 + l16;
    Mm[i * OO + j] = (i >= j) ? c[v] * invN : 0.0f;
  }
}

// ---------------------------------------------------------------------------
// 6) delta_w = A - M @ Wf       [O, D], K = O = 256
// ---------------------------------------------------------------------------
__global__ void dgemm_k(const float* __restrict__ Mm, const float* __restrict__ W,
                        const float* __restrict__ Aw, float* __restrict__ dw) {
  int lane = threadIdx.x & 31;
  int wid  = threadIdx.x >> 5;
  int tile = blockIdx.x * 8 + wid;      // 1600 tiles = 16 (O) x 100 (D)
  int tO   = tile / 100;
  int tD   = tile - tO * 100;
  int half = lane >> 4;
  int l16  = lane & 15;

  const float* ap = Mm + (long)(tO * 16 + l16) * OO + 2 * half;  // A[m][k]=M[o][o']
  const float* bp = W  + (long)(2 * half) * DD + tD * 16 + l16;  // B[k][n]=Wf[o'][d]

  v8f c = {};
#pragma unroll 4
  for (int k = 0; k < OO; k += 4) {
    v2f a = *(const v2f*)ap;
    v2f b; b.x = bp[0]; b.y = bp[DD];
    c = wmma_f32(a, b, c);
    ap += 4; bp += 4 * DD;
  }

#pragma unroll
  for (int v = 0; v < 8; ++v) {
    int o = tO * 16 + v + 8 * half;
    int d = tD * 16 + l16;
    dw[(long)o * DD + d] = Aw[(long)o * DD + d] - c[v];
  }
}

// ---------------------------------------------------------------------------
// launcher
// ---------------------------------------------------------------------------
extern "C" void kernel_launch(void* const* d_in, const int* in_sizes, int n_in,
                              void* d_out, int out_size, void* d_ws, size_t ws_size,
                              hipStream_t stream) {
  const float* x = (const float*)d_in[0];    // [32,64,32,32]
  const float* W = (const float*)d_in[1];    // [256,64,5,5] == Wf[256,1600]

  float* out_y4 = (float*)d_out;             // first Y4SZ floats
  float* out_dw = out_y4 + Y4SZ;             // then O*D floats

  // workspace layout (floats): xu | y(=r in place) | Aw | Mm
  float* xu = (float*)d_ws;                                 // NN*DD   = 40,140,800
  float* yr = xu + (size_t)NN * DD;                         // NN*OO   =  6,422,528
  float* Aw = yr + (size_t)NN * OO;                         // OO*DD   =    409,600
  float* Mm = Aw + (size_t)OO * DD;                         // OO*OO   =     65,536

  im2col_k<<<NN, 256, 0, stream>>>(x, xu);
  fwd_k  <<<NN / 8, 256, 0, stream>>>(xu, W, yr, out_y4);   // 25088 tiles / 8 waves
  lfb_k  <<<NN, 256, 0, stream>>>(yr);                      // yr: y -> r in place
  agemm_k<<<200, 256, 0, stream>>>(yr, xu, Aw);             // 1600 tiles
  mgemm_k<<<32, 256, 0, stream>>>(yr, Mm);                  //  256 tiles
  dgemm_k<<<200, 256, 0, stream>>>(Mm, W, Aw, out_dw);      // 1600 tiles
}